// CurrentFactorCell_18167711662678
// MI455X (gfx1250) — compile-verified
//
#include <hip/hip_runtime.h>

// Elementwise complex gate-mix over N=8.4M elements.
// Traffic: 24 B/elem -> ~201 MB total; at 23.3 TB/s the floor is ~8.6 us.
// Flops: ~16/elem -> ~130 MFLOP, negligible. Pure HBM-bound streaming:
// branchless B128 NT load/store hot kernel, separate tail kernel (host-gated).
// No WMMA/TDM: no matrix structure or reuse to amortize LDS staging.

typedef float v4f __attribute__((ext_vector_type(4)));

__global__ __launch_bounds__(256) void gatecell_stream_kernel(
    const float* __restrict__ z_re,
    const float* __restrict__ z_im,
    const float* __restrict__ gate,   // (N,2) interleaved
    const float* __restrict__ scale,  // [2]
    const float* __restrict__ mix,    // [2]
    const float* __restrict__ bias,   // [2]
    float* __restrict__ out,          // [2N]: out_re then out_im
    int n_groups,                     // number of 8-element groups
    int n) {
  const int g = blockIdx.x * blockDim.x + threadIdx.x;
  if (g >= n_groups) return;  // only the last partial block's lanes exit
  const size_t i = (size_t)g * 8;

  // Wave-uniform coefficients -> s_load + scalar math.
  const float s0 = scale[0], s1 = scale[1];
  const float m0 = mix[0],   m1 = mix[1];
  const float c0 = bias[0],  c1 = bias[1];
  // out_re = (s0+m0)*(zr*gr) + (m0-s0)*(zi*gi) + c0
  // out_im = (s1-m1)*(zr*gi) + (s1+m1)*(zi*gr) + c1
  const float a_rr = s0 + m0;
  const float a_ii = m0 - s0;
  const float a_ri = s1 - m1;
  const float a_ir = s1 + m1;

  // 8 elements: all memory ops are global_*_b128 with NT temporal hint.
  v4f zr0 = __builtin_nontemporal_load((const v4f*)(z_re + i));
  v4f zr1 = __builtin_nontemporal_load((const v4f*)(z_re + i + 4));
  v4f zi0 = __builtin_nontemporal_load((const v4f*)(z_im + i));
  v4f zi1 = __builtin_nontemporal_load((const v4f*)(z_im + i + 4));
  v4f g0  = __builtin_nontemporal_load((const v4f*)(gate + 2 * i));
  v4f g1  = __builtin_nontemporal_load((const v4f*)(gate + 2 * i + 4));
  v4f g2  = __builtin_nontemporal_load((const v4f*)(gate + 2 * i + 8));
  v4f g3  = __builtin_nontemporal_load((const v4f*)(gate + 2 * i + 12));

  // Deinterleave gate pairs (register swizzles).
  v4f gr0 = (v4f){g0.x, g0.z, g1.x, g1.z};
  v4f gi0 = (v4f){g0.y, g0.w, g1.y, g1.w};
  v4f gr1 = (v4f){g2.x, g2.z, g3.x, g3.z};
  v4f gi1 = (v4f){g2.y, g2.w, g3.y, g3.w};

  // 2 mul + 2 fma per output lane (contracted to v_fma_f32).
  v4f ore0 = (zr0 * gr0) * a_rr + ((zi0 * gi0) * a_ii + c0);
  v4f ore1 = (zr1 * gr1) * a_rr + ((zi1 * gi1) * a_ii + c0);
  v4f oim0 = (zr0 * gi0) * a_ri + ((zi0 * gr0) * a_ir + c1);
  v4f oim1 = (zr1 * gi1) * a_ri + ((zi1 * gr1) * a_ir + c1);

  float* out_re = out;
  float* out_im = out + (size_t)n;
  __builtin_nontemporal_store(ore0, (v4f*)(out_re + i));
  __builtin_nontemporal_store(ore1, (v4f*)(out_re + i + 4));
  __builtin_nontemporal_store(oim0, (v4f*)(out_im + i));
  __builtin_nontemporal_store(oim1, (v4f*)(out_im + i + 4));
}

// Remainder elements (n % 8) — launched only when needed (never for N=8M).
__global__ __launch_bounds__(64) void gatecell_tail_kernel(
    const float* __restrict__ z_re,
    const float* __restrict__ z_im,
    const float* __restrict__ gate,
    const float* __restrict__ scale,
    const float* __restrict__ mix,
    const float* __restrict__ bias,
    float* __restrict__ out,
    int start, int n) {
  const int k = start + blockIdx.x * blockDim.x + threadIdx.x;
  if (k >= n) return;
  const float s0 = scale[0], s1 = scale[1];
  const float m0 = mix[0],   m1 = mix[1];
  const float c0 = bias[0],  c1 = bias[1];
  const float a_rr = s0 + m0, a_ii = m0 - s0;
  const float a_ri = s1 - m1, a_ir = s1 + m1;
  float zr = z_re[k], zi = z_im[k];
  float gr = gate[2 * (size_t)k], gi = gate[2 * (size_t)k + 1];
  out[k]             = (zr * gr) * a_rr + ((zi * gi) * a_ii + c0);
  out[(size_t)n + k] = (zr * gi) * a_ri + ((zi * gr) * a_ir + c1);
}

extern "C" void kernel_launch(void* const* d_in, const int* in_sizes, int n_in,
                              void* d_out, int out_size, void* d_ws, size_t ws_size,
                              hipStream_t stream) {
  const float* z_re  = (const float*)d_in[0];
  const float* z_im  = (const float*)d_in[1];
  const float* gate  = (const float*)d_in[2];
  const float* scale = (const float*)d_in[3];
  const float* mix   = (const float*)d_in[4];
  const float* bias  = (const float*)d_in[5];
  float* out = (float*)d_out;

  const int n = in_sizes[0];        // 8,388,608
  const int n_groups = n / 8;       // full 8-element groups
  const int block = 256;            // 8 wave32 per block
  const int grid = (n_groups + block - 1) / block;

  if (n_groups > 0) {
    hipLaunchKernelGGL(gatecell_stream_kernel, dim3(grid), dim3(block), 0,
                       stream, z_re, z_im, gate, scale, mix, bias, out,
                       n_groups, n);
  }
  const int tail_start = n_groups * 8;
  if (tail_start < n) {  // not taken for N = 8M (deterministic in n)
    const int tail = n - tail_start;
    hipLaunchKernelGGL(gatecell_tail_kernel, dim3((tail + 63) / 64), dim3(64),
                       0, stream, z_re, z_im, gate, scale, mix, bias, out,
                       tail_start, n);
  }
}